// OuterProductPair_21088289423633
// MI455X (gfx1250) — compile-verified
//
#include <hip/hip_runtime.h>
#include <hip/hip_bf16.h>

typedef __attribute__((ext_vector_type(2))) float v2f;
typedef __attribute__((ext_vector_type(8))) float v8f;

#define L_DIM 512
#define C_DIM 768

// ---------------------------------------------------------------------------
// Kernel A: LayerNorm + W1 GEMV + chunk-averaged outer product + Wa GEMV+gelu
// One workgroup per (b,l).  Produces mean_out / mean_trans in [B,32,L,64].
// ---------------------------------------------------------------------------
__global__ __launch_bounds__(256) void fused_front_kernel(
    const float* __restrict__ x,
    const float* __restrict__ ln_w, const float* __restrict__ ln_b,
    const float* __restrict__ W1, const float* __restrict__ b1,
    const float* __restrict__ Wa, const float* __restrict__ ba,
    float* __restrict__ mo, float* __restrict__ mt)
{
    __shared__ float hbuf[C_DIM];
    __shared__ float tbuf[C_DIM];      // 3 x 256
    __shared__ float mobuf[32 * 64];
    __shared__ float red[16];

    const int bl   = blockIdx.x;       // 0..B*L-1
    const int tid  = threadIdx.x;
    const int lane = tid & 31;
    const int wid  = tid >> 5;

    const float* xrow = x + (size_t)bl * C_DIM;
    float v0 = xrow[tid];
    float v1 = xrow[tid + 256];
    float v2 = xrow[tid + 512];

    // mean
    float s = v0 + v1 + v2;
    #pragma unroll
    for (int off = 16; off > 0; off >>= 1) s += __shfl_down(s, off, 32);
    if (lane == 0) red[wid] = s;
    __syncthreads();
    if (tid == 0) {
        float t = 0.f;
        #pragma unroll
        for (int w = 0; w < 8; ++w) t += red[w];
        red[8] = t * (1.0f / (float)C_DIM);
    }
    __syncthreads();
    const float mu = red[8];

    // variance
    float d0 = v0 - mu, d1 = v1 - mu, d2 = v2 - mu;
    float sv = d0 * d0 + d1 * d1 + d2 * d2;
    #pragma unroll
    for (int off = 16; off > 0; off >>= 1) sv += __shfl_down(sv, off, 32);
    if (lane == 0) red[wid] = sv;
    __syncthreads();
    if (tid == 0) {
        float t = 0.f;
        #pragma unroll
        for (int w = 0; w < 8; ++w) t += red[w];
        red[9] = rsqrtf(t * (1.0f / (float)C_DIM) + 1e-6f);
    }
    __syncthreads();
    const float rstd = red[9];

    hbuf[tid]       = d0 * rstd * ln_w[tid]       + ln_b[tid];
    hbuf[tid + 256] = d1 * rstd * ln_w[tid + 256] + ln_b[tid + 256];
    hbuf[tid + 512] = d2 * rstd * ln_w[tid + 512] + ln_b[tid + 512];
    __syncthreads();

    // t[k][o] = b1[o] + sum_i W1[o][i] * h[k*256+i]   (thread owns output o=tid)
    {
        const float* wrow = W1 + (size_t)tid * 256;
        float a0 = b1[tid];
        float a1 = a0, a2 = a0;
        #pragma unroll 4
        for (int i = 0; i < 256; ++i) {
            float w = wrow[i];
            a0 = fmaf(w, hbuf[i],       a0);
            a1 = fmaf(w, hbuf[i + 256], a1);
            a2 = fmaf(w, hbuf[i + 512], a2);
        }
        tbuf[tid]       = a0;
        tbuf[tid + 256] = a1;
        tbuf[tid + 512] = a2;
    }
    __syncthreads();

    const int b = bl / L_DIM;
    const int l = bl % L_DIM;

    // mean over 3 chunks of outer products: [32][8][8] -> [32][64]
    #pragma unroll
    for (int r = 0; r < 8; ++r) {
        int idx = r * 256 + tid;           // 0..2047
        int c   = idx >> 6;                // 0..31
        int ij  = idx & 63;
        int i   = ij >> 3, j = ij & 7;
        float val = (tbuf[c * 8 + i]       * tbuf[c * 8 + j] +
                     tbuf[256 + c * 8 + i] * tbuf[256 + c * 8 + j] +
                     tbuf[512 + c * 8 + i] * tbuf[512 + c * 8 + j]) / 3.0f;
        mobuf[idx] = val;
        mo[(((size_t)b * 32 + c) * L_DIM + l) * 64 + ij] = val;
    }
    __syncthreads();

    // mean_trans = gelu(mean_out @ Wa^T + ba), exact gelu (erf)
    #pragma unroll
    for (int r = 0; r < 8; ++r) {
        int idx = r * 256 + tid;
        int c   = idx >> 6;
        int d   = idx & 63;
        const float* wr = Wa + d * 64;
        const float* mrow = &mobuf[c * 64];
        float acc = ba[d];
        #pragma unroll 4
        for (int e = 0; e < 64; ++e) acc = fmaf(wr[e], mrow[e], acc);
        float g = 0.5f * acc * (1.0f + erff(acc * 0.70710678118654752f));
        mt[(((size_t)b * 32 + c) * L_DIM + l) * 64 + d] = g;
    }
}

// ---------------------------------------------------------------------------
// Kernel B: per 16x16 (l,m) tile:
//   stage 1: raw_c = mean_out_tile(16x64) @ mean_trans_tile(16x64)^T  (f32 WMMA)
//   stage 2: out(256x128) = raw(256x32) @ W2^T(32x128) + b2           (f32 WMMA)
// 8 waves / 256 threads per workgroup.
// ---------------------------------------------------------------------------
#define RAW_STRIDE 33   // pad to dodge LDS bank conflicts

__global__ __launch_bounds__(256) void pair_back_kernel(
    const float* __restrict__ mo, const float* __restrict__ mt,
    const float* __restrict__ W2, const float* __restrict__ b2,
    float* __restrict__ out)
{
    __shared__ float raw[256 * RAW_STRIDE];   // [lm=256][c=32] padded

    const int bidx  = blockIdx.x;             // 0..2047
    const int b     = bidx >> 10;
    const int lt    = (bidx >> 5) & 31;
    const int mtile = bidx & 31;
    const int l0 = lt * 16, m0 = mtile * 16;

    const int tid   = threadIdx.x;
    const int lane  = tid & 31;
    const int wid   = tid >> 5;
    const int lsel  = lane & 15;              // M (stage1) / N row-sel per layout
    const int khalf = lane >> 4;              // selects K pair {4s+2,4s+3}

    // ---- stage 1: each wave computes 4 of the 32 channels ----
    #pragma unroll
    for (int cc = 0; cc < 4; ++cc) {
        const int c = wid * 4 + cc;
        const float* A  = mo + (((size_t)b * 32 + c) * L_DIM + l0) * 64;
        const float* Bt = mt + (((size_t)b * 32 + c) * L_DIM + m0) * 64;
        const float* aptr = A  + lsel * 64 + khalf * 2;  // A[M=lsel][4s+2*khalf+{0,1}]
        const float* bptr = Bt + lsel * 64 + khalf * 2;  // B[k][N=lsel] = mt[m0+N][k]
        v8f acc = {0.f, 0.f, 0.f, 0.f, 0.f, 0.f, 0.f, 0.f};
        #pragma unroll
        for (int s = 0; s < 16; ++s) {                   // K = 64 = 16 * 4
            v2f a  = *(const v2f*)(aptr + s * 4);
            v2f bb = *(const v2f*)(bptr + s * 4);
            acc = __builtin_amdgcn_wmma_f32_16x16x4_f32(
                false, a, false, bb, (short)0, acc, false, false);
        }
        // D layout: vgpr r, lane L -> M = r + 8*(L/16), N = L%16
        #pragma unroll
        for (int r = 0; r < 8; ++r) {
            int lm = (r + 8 * khalf) * 16 + lsel;        // l-local*16 + m-local
            raw[lm * RAW_STRIDE + c] = acc[r];
        }
    }
    __syncthreads();

    // ---- stage 2: [256 x 32] @ W2^T [32 x 128] + b2; 128 tiles, 16 per wave ----
    for (int t = 0; t < 16; ++t) {
        const int tileId = wid * 16 + t;
        const int lmBase = (tileId >> 3) * 16;           // 16 lm-tiles
        const int pBase  = (tileId & 7) * 16;            // 8 p-tiles
        const float bias = b2[pBase + lsel];             // bias depends only on N
        v8f acc;
        #pragma unroll
        for (int r = 0; r < 8; ++r) acc[r] = bias;

        const float* arow = &raw[(lmBase + lsel) * RAW_STRIDE + khalf * 2];
        const float* brow = W2 + (size_t)(pBase + lsel) * 32 + khalf * 2; // B[k][N]=W2[p][k]
        #pragma unroll
        for (int s = 0; s < 8; ++s) {                    // K = 32 = 8 * 4
            v2f a;
            a.x = arow[s * 4];
            a.y = arow[s * 4 + 1];
            v2f bb = *(const v2f*)(brow + s * 4);
            acc = __builtin_amdgcn_wmma_f32_16x16x4_f32(
                false, a, false, bb, (short)0, acc, false, false);
        }
        // store: vgpr r, lane L -> lm = lmBase + r + 8*khalf, p = pBase + lsel
        #pragma unroll
        for (int r = 0; r < 8; ++r) {
            int lm = lmBase + r + 8 * khalf;
            int l  = l0 + (lm >> 4);
            int m  = m0 + (lm & 15);
            out[(((size_t)b * L_DIM + l) * L_DIM + m) * 128 + pBase + lsel] = acc[r];
        }
    }
}

// ---------------------------------------------------------------------------
extern "C" void kernel_launch(void* const* d_in, const int* in_sizes, int n_in,
                              void* d_out, int out_size, void* d_ws, size_t ws_size,
                              hipStream_t stream) {
    const float* x    = (const float*)d_in[0];
    const float* ln_w = (const float*)d_in[1];
    const float* ln_b = (const float*)d_in[2];
    const float* W1   = (const float*)d_in[3];
    const float* b1   = (const float*)d_in[4];
    const float* Wa   = (const float*)d_in[5];
    const float* ba   = (const float*)d_in[6];
    const float* W2   = (const float*)d_in[7];
    const float* b2   = (const float*)d_in[8];
    float* out = (float*)d_out;

    float* mo = (float*)d_ws;                                   // [2,32,512,64] = 8 MB
    float* mt = mo + (size_t)2 * 32 * L_DIM * 64;               // [2,32,512,64] = 8 MB

    fused_front_kernel<<<2 * L_DIM, 256, 0, stream>>>(x, ln_w, ln_b, W1, b1, Wa, ba, mo, mt);
    pair_back_kernel<<<2 * 32 * 32, 256, 0, stream>>>(mo, mt, W2, b2, out);
}